// FastMaskedDense1D_82678120448545
// MI455X (gfx1250) — compile-verified
//
#include <hip/hip_runtime.h>

typedef __attribute__((ext_vector_type(2))) float v2f;
typedef __attribute__((ext_vector_type(4))) float v4f;
typedef __attribute__((ext_vector_type(8))) float v8f;
typedef __attribute__((ext_vector_type(4))) unsigned int v4u;
typedef __attribute__((ext_vector_type(4))) int v4i;
typedef __attribute__((ext_vector_type(8))) int v8i;

namespace {
constexpr int SIZE_ = 512;
constexpr int FEAT  = 16;             // output features (N)
constexpr int INF   = 16;             // input features
constexpr int BATCH = 8192;           // M
constexpr int EXCL  = 1;
constexpr int KROW  = SIZE_ * FEAT;   // 8192 floats: kernel-matrix row stride
constexpr int CROW  = SIZE_ * INF;    // 8192 floats: cache per-batch stride
constexpr int CH    = 16;             // sites (j) per TDM chunk
constexpr int CHUNK_FLOATS = CH * 16 * 16;     // 4096 floats per A or B tile
constexpr int PARITY_FLOATS = 2 * CHUNK_FLOATS; // A+B for one buffer parity
constexpr unsigned LDS_BYTES = 2 * PARITY_FLOATS * 4; // double buffered: 64 KB
}

extern __shared__ float smem[];   // dynamic LDS: [A0|B0|A1|B1], 16K floats

// ---------------------------------------------------------------------------
// TDM descriptor helpers (D# layout per CDNA5 ISA).
// Group1 is constant per tensor shape; group0 carries lds/global addresses.
// ---------------------------------------------------------------------------
__device__ __forceinline__ v8i tdm_group1(unsigned td0, unsigned td1,
                                          unsigned tile0, unsigned tile1,
                                          unsigned long long stride0)
{
    v8i g;
    g[0] = (int)(2u << 16);                                  // data_size=4B
    g[1] = (int)((td0 & 0xFFFFu) << 16);                     // tensor_dim0 lo
    g[2] = (int)((td0 >> 16) | ((td1 & 0xFFFFu) << 16));     // td0 hi | td1 lo
    g[3] = (int)((td1 >> 16) | (tile0 << 16));               // td1 hi | tile0
    g[4] = (int)(tile1 & 0xFFFFu);                           // tile1, tile2=0
    g[5] = (int)(stride0 & 0xFFFFFFFFull);                   // dim0 stride lo
    g[6] = (int)((stride0 >> 32) & 0xFFFFull);               // stride hi, s1=0
    g[7] = 0;
    return g;
}

__device__ __forceinline__ void tdm_issue(unsigned lds_byte_off,
                                          const float* gaddr, v8i g1)
{
    unsigned long long ga = (unsigned long long)(size_t)gaddr;
    v4u g0;
    g0[0] = 1u;                                   // count=1, user descriptor
    g0[1] = lds_byte_off;                         // LDS byte address of tile
    g0[2] = (unsigned)ga;                         // global_addr[31:0]
    g0[3] = (unsigned)(ga >> 32) | (2u << 30);    // global_addr[56:32]|type=2
    v4i z4 = {0, 0, 0, 0};                        // groups 2/3: 2D tensor
    v8i z8 = {0, 0, 0, 0, 0, 0, 0, 0};
    // amdgpu-toolchain (clang-23) 6-arg form.
    __builtin_amdgcn_tensor_load_to_lds(g0, g1, z4, z4, z8, 0);
}

// ---------------------------------------------------------------------------
// 4x V_WMMA_F32_16X16X4_F32 covering one 16-wide K-slab.
// K-permutation per half-wave: lanes 0-15 own K-slots {0,1}, lanes 16-31 own
// {2,3}; A and B use the identical permutation so the dot product is exact.
// ---------------------------------------------------------------------------
__device__ __forceinline__ v8f wmma4(v8f acc, v4f a0, v4f a1,
                                     float b00, float b01, float b10, float b11,
                                     float b20, float b21, float b30, float b31)
{
    v2f a, b;
    a[0] = a0[0]; a[1] = a0[1]; b[0] = b00; b[1] = b01;
    acc = __builtin_amdgcn_wmma_f32_16x16x4_f32(false, a, false, b,
                                                (short)0, acc, false, false);
    a[0] = a0[2]; a[1] = a0[3]; b[0] = b10; b[1] = b11;
    acc = __builtin_amdgcn_wmma_f32_16x16x4_f32(false, a, false, b,
                                                (short)0, acc, false, false);
    a[0] = a1[0]; a[1] = a1[1]; b[0] = b20; b[1] = b21;
    acc = __builtin_amdgcn_wmma_f32_16x16x4_f32(false, a, false, b,
                                                (short)0, acc, false, false);
    a[0] = a1[2]; a[1] = a1[3]; b[0] = b30; b[1] = b31;
    acc = __builtin_amdgcn_wmma_f32_16x16x4_f32(false, a, false, b,
                                                (short)0, acc, false, false);
    return acc;
}

__global__ __launch_bounds__(32)
void fmd1d_update_site(const float* __restrict__ inputs,
                       const float* __restrict__ cache,
                       const float* __restrict__ kernelm,
                       const float* __restrict__ bias,
                       const int*   __restrict__ idxp,
                       float*       __restrict__ out)
{
    const int index = idxp[0];          // uniform scalar load from device mem
    const int jlim  = index - EXCL;     // last contributing site (from inputs)
    const int col0  = index * FEAT;     // kernel column block of this site

    const int lane = threadIdx.x & 31;
    const int m    = lane & 15;         // A-row-in-tile, B/C column
    const int half = lane >> 4;
    const int tile = blockIdx.x;        // one wave32 workgroup per 16-row tile
    const int row  = tile * 16 + m;

    // Constant D# group1 descriptors:
    //  A tile: 16 batch rows (dim1, stride 8192) x CH*16 contiguous floats.
    //  B tile: CH*16 kernel rows (dim1, stride 8192) x 16 contiguous floats.
    const v8i g1A = tdm_group1(CH * 16, 16, CH * 16, 16,
                               (unsigned long long)CROW);
    const v8i g1B = tdm_group1(16, CH * 16, 16, CH * 16,
                               (unsigned long long)KROW);

    const unsigned lds_base = (unsigned)(size_t)(void*)smem; // LDS byte offset
    const float* a_base = cache + (size_t)tile * 16 * CROW;  // tile row 0
    const float* b_base = kernelm + col0;                    // kernel_i col 0

    const int nchunks = (jlim > 0) ? (jlim + CH - 1) / CH : 0;

    // Prologue: issue chunk 0 (A + B tensor loads; TENSORcnt += 2).
    if (nchunks > 0) {
        tdm_issue(lds_base, a_base, g1A);
        tdm_issue(lds_base + CHUNK_FLOATS * 4,
                  b_base + (size_t)0 * KROW, g1B);
    }

    v8f acc = {};

    for (int c = 0; c < nchunks; ++c) {
        const int p = c & 1;
        if (c + 1 < nchunks) {
            // The buffer we are about to overwrite was last read two chunks
            // ago; make sure those ds_loads have landed before TDM rewrites.
            asm volatile("s_wait_dscnt 0x0" ::: "memory");
            const unsigned pb = (unsigned)((c + 1) & 1) * PARITY_FLOATS * 4;
            tdm_issue(lds_base + pb,
                      a_base + (size_t)(c + 1) * CH * INF, g1A);
            tdm_issue(lds_base + pb + CHUNK_FLOATS * 4,
                      b_base + (size_t)(c + 1) * CH * INF * KROW, g1B);
            // Retire the two oldest tensor ops (= chunk c, FIFO per wave).
            asm volatile("s_wait_tensorcnt 0x2" ::: "memory");
        } else {
            asm volatile("s_wait_tensorcnt 0x0" ::: "memory");
        }

        const float* Ab = smem + p * PARITY_FLOATS;
        const float* Bb = Ab + CHUNK_FLOATS;
        const int jrem = jlim - c * CH;
        const int jcnt = (jrem < CH) ? jrem : CH;

        #pragma unroll 4
        for (int jl = 0; jl < jcnt; ++jl) {
            // A: lane m reads its batch row; two ds_load_b128.
            const float* as = Ab + m * (CH * 16) + jl * 16 + half * 4;
            v4f a0 = *(const v4f*)as;
            v4f a1 = *(const v4f*)(as + 8);
            // B: paired offsets -> ds_load_2addr_b32 candidates.
            const float* bs = Bb + (jl * 16 + half * 4) * 16 + m;
            acc = wmma4(acc, a0, a1,
                        bs[0],   bs[16],  bs[32],  bs[48],
                        bs[128], bs[144], bs[160], bs[176]);
        }
    }

    // Site j == jlim is the freshly scattered `inputs` row (the cache input
    // buffer is never mutated; we substitute the source pointer instead).
    if (jlim >= 0) {
        const float* src = inputs + (size_t)row * INF;
        v4f a0 = *(const v4f*)(src + half * 4);
        v4f a1 = *(const v4f*)(src + 8 + half * 4);
        const float* bh = kernelm + ((size_t)jlim * INF + half * 4) * KROW
                        + col0 + m;
        const float* bh2 = bh + (size_t)8 * KROW;
        acc = wmma4(acc, a0, a1,
                    bh[0],  bh[KROW],  bh[2 * KROW],  bh[3 * KROW],
                    bh2[0], bh2[KROW], bh2[2 * KROW], bh2[3 * KROW]);
    }

    // C/D layout: VGPR i holds (M = i + half*8, N = m). Add bias[index, m];
    // each half-wave writes contiguous 64B output rows.
    const float bv = bias[col0 + m];
    float* orow = out + (size_t)(tile * 16 + half * 8) * FEAT + m;
    #pragma unroll
    for (int i = 0; i < 8; ++i)
        orow[(size_t)i * FEAT] = acc[i] + bv;
}

extern "C" void kernel_launch(void* const* d_in, const int* in_sizes, int n_in,
                              void* d_out, int out_size, void* d_ws, size_t ws_size,
                              hipStream_t stream)
{
    const float* inputs  = (const float*)d_in[0];
    const float* cache   = (const float*)d_in[1];
    const float* kernelm = (const float*)d_in[2];
    const float* bias    = (const float*)d_in[3];
    const int*   idxp    = (const int*)d_in[4];
    float* out = (float*)d_out;

    // One wave32 workgroup per 16-row M-tile; 64 KB dynamic LDS each
    // (double-buffered A+B TDM chunks). 512 blocks spread across WGPs.
    const int tiles = BATCH / 16;  // 512
    fmd1d_update_site<<<tiles, 32, LDS_BYTES, stream>>>(inputs, cache, kernelm,
                                                        bias, idxp, out);
}